// NetAE_35038343200962
// MI455X (gfx1250) — compile-verified
//
#include <hip/hip_runtime.h>

// ---------------------------------------------------------------------------
// MMoE-style net, batch=1: a chain of fp32 GEMVs. Bandwidth-bound (AI ~0.5
// FLOP/B, ~604 MB of single-use fp32 weights -> ~26 us floor at 23.3 TB/s).
// GEMV runs on V_WMMA_F32_16X16X4_F32: A = input vector replicated across the
// 16 rows, B = 4x16 tile of W, so every row of D holds the same 16 partial
// dot products. Weights stream with non-temporal loads; K-loop unrolled x4
// for memory-level parallelism.
// ---------------------------------------------------------------------------

typedef float v2f __attribute__((ext_vector_type(2)));
typedef float v8f __attribute__((ext_vector_type(8)));

#define D_IN   2048
#define D_H    2048
#define D_V    2048
#define D_A    256
#define N_EXP  16
#define N_TASK 10
#define D_T    2048
#define D_OUT  512

// y[b][j] = act( sum_i x[b][i] * W[b][i*N + j] + bias[b][j] )
//   W: [batch, K, N] row-major, K multiple of 128, N multiple of 32.
//   grid = (N/32, batch), block = 256 (8 wave32s; waves split K, LDS-reduce).
__global__ __launch_bounds__(256) void gemv_wmma(
    const float* __restrict__ x, const float* __restrict__ W,
    const float* __restrict__ bias, float* __restrict__ y,
    int K, int N, int xBatchStride, int doRelu)
{
    __shared__ float xs[2048];       // staged input vector (K <= 2048 here)
    __shared__ float partial[8 * 32];

    const int tid   = threadIdx.x;
    const int lane  = tid & 31;
    const int wave  = tid >> 5;
    const int batch = blockIdx.y;
    const int j0    = blockIdx.x * 32;

    const float* xb = x + (size_t)batch * xBatchStride;
    const float* Wb = W + (size_t)batch * K * N;

    for (int i = tid; i < K; i += 256) xs[i] = xb[i];
    __syncthreads();

    const int Kc   = K >> 3;               // per-wave K chunk
    const int iBeg = wave * Kc;
    // WMMA f32 16x16x4 per-lane K slot: VGPR0 holds K=0 (lanes 0-15) / K=2
    // (lanes 16-31); VGPR1 holds K=1 / K=3.  Same split for A and B, so each
    // lane pairs x[i0+k] with W[i0+k][col] for matching k.
    const int rsel = (lane < 16) ? 0 : 2;
    const int colA = j0 + (lane & 15);

    // per-lane streaming pointers for the two 16-wide column tiles
    const float* WpA = Wb + (size_t)(iBeg + rsel) * N + colA;
    const float* WpB = WpA + 16;
    const float2* xp = (const float2*)(xs + iBeg + rsel);   // ds_load_b64/step

    v8f c0 = {};
    v8f c1 = {};
    const int steps = Kc >> 2;              // K-steps of 4
    #pragma unroll 4
    for (int s = 0; s < steps; ++s) {
        const float2 xv = xp[0];
        v2f a;                              // A[m][k] = x[i0+k] for every m
        a[0] = xv.x;
        a[1] = xv.y;
        v2f bA, bB;                         // B[k][n] = W[i0+k][j0+n] (NT stream)
        bA[0] = __builtin_nontemporal_load(WpA);
        bA[1] = __builtin_nontemporal_load(WpA + N);
        bB[0] = __builtin_nontemporal_load(WpB);
        bB[1] = __builtin_nontemporal_load(WpB + N);
        c0 = __builtin_amdgcn_wmma_f32_16x16x4_f32(false, a, false, bA,
                                                   (short)0, c0, false, false);
        c1 = __builtin_amdgcn_wmma_f32_16x16x4_f32(false, a, false, bB,
                                                   (short)0, c1, false, false);
        WpA += 4 * (size_t)N;
        WpB += 4 * (size_t)N;
        xp  += 2;                           // 4 floats
    }

    // Every row of D is identical; row 0 lives in acc[0], lanes 0-15 (lanes
    // 16-31 duplicate it as row M=8).  Publish this wave's 32 partials.
    if (lane < 16) {
        partial[wave * 32 + lane]      = c0[0];
        partial[wave * 32 + 16 + lane] = c1[0];
    }
    __syncthreads();

    if (tid < 32) {                         // reduce the 8 K-chunks
        float s = 0.f;
        for (int w = 0; w < 8; ++w) s += partial[w * 32 + tid];
        s += bias[(size_t)batch * N + j0 + tid];
        if (doRelu) s = fmaxf(s, 0.f);
        y[(size_t)batch * N + j0 + tid] = s;
    }
}

// q = W_q[task][task,:] + b_q[task];  ek[n] = k[n] . q;  res[j] = sum_n ek[n]*v[n][j]
// grid = D_V/256 blocks; each block redundantly computes q/ek (tiny) then owns
// a 256-wide slice of res.  Fully deterministic.
__global__ __launch_bounds__(256) void attend_combine(
    const float* __restrict__ kk, const float* __restrict__ v,
    const float* __restrict__ W_q, const float* __restrict__ b_q,
    const int* __restrict__ task_id_p, float* __restrict__ res)
{
    __shared__ float q[D_A];
    __shared__ float red[256];
    __shared__ float ek[N_EXP];

    const int tid = threadIdx.x;
    const int t   = task_id_p[0];

    // one-hot(task) @ W_q[task] selects row `t` of W_q[task]
    q[tid] = W_q[((size_t)t * N_TASK + t) * D_A + tid] + b_q[(size_t)t * D_A + tid];
    __syncthreads();

    const int g = tid >> 4, l = tid & 15;   // 16 experts x 16 threads
    float p = 0.f;
    for (int a = l; a < D_A; a += 16) p += kk[g * D_A + a] * q[a];
    red[tid] = p;
    __syncthreads();
    if (tid < N_EXP) {
        float s = 0.f;
        for (int i = 0; i < 16; ++i) s += red[tid * 16 + i];
        ek[tid] = s;
    }
    __syncthreads();

    const int j = blockIdx.x * 256 + tid;
    float s = 0.f;
    for (int n = 0; n < N_EXP; ++n) s += ek[n] * v[(size_t)n * D_V + j];
    res[j] = s;
}

extern "C" void kernel_launch(void* const* d_in, const int* in_sizes, int n_in,
                              void* d_out, int out_size, void* d_ws, size_t ws_size,
                              hipStream_t stream) {
    (void)in_sizes; (void)n_in; (void)out_size; (void)ws_size;
    const float* x     = (const float*)d_in[0];
    const float* W1    = (const float*)d_in[1];
    const float* b1    = (const float*)d_in[2];
    const float* W2    = (const float*)d_in[3];
    const float* b2    = (const float*)d_in[4];
    const float* W_exp = (const float*)d_in[5];
    const float* b_exp = (const float*)d_in[6];
    const float* W_v   = (const float*)d_in[7];
    const float* b_v   = (const float*)d_in[8];
    const float* W_k   = (const float*)d_in[9];
    const float* b_k   = (const float*)d_in[10];
    const float* W_q   = (const float*)d_in[11];
    const float* b_q   = (const float*)d_in[12];
    const float* W_t   = (const float*)d_in[13];
    const float* b_t   = (const float*)d_in[14];
    const float* W_l   = (const float*)d_in[15];
    const float* b_l   = (const float*)d_in[16];
    const int*   tid   = (const int*)d_in[17];
    float* out = (float*)d_out;

    float* ws  = (float*)d_ws;              // fp32 scratch layout (elements):
    float* h1  = ws;                        // [2048]
    float* h2  = ws + 2048;                 // [2048]
    float* e   = ws + 4096;                 // [16,2048]
    float* vv  = ws + 36864;                // [16,2048]
    float* kk  = ws + 69632;                // [16,256]
    float* res = ws + 73728;                // [2048]
    float* tw  = ws + 75776;                // [2048]

    dim3 blk(256);
    gemv_wmma<<<dim3(D_H  / 32, 1),     blk, 0, stream>>>(x,  W1,    b1,    h1, D_IN, D_H,  0,   1);
    gemv_wmma<<<dim3(D_H  / 32, 1),     blk, 0, stream>>>(h1, W2,    b2,    h2, D_H,  D_H,  0,   1);
    gemv_wmma<<<dim3(D_H  / 32, N_EXP), blk, 0, stream>>>(h2, W_exp, b_exp, e,  D_H,  D_H,  0,   1);
    gemv_wmma<<<dim3(D_V  / 32, N_EXP), blk, 0, stream>>>(e,  W_v,   b_v,   vv, D_H,  D_V,  D_H, 0);
    gemv_wmma<<<dim3(D_A  / 32, N_EXP), blk, 0, stream>>>(e,  W_k,   b_k,   kk, D_H,  D_A,  D_H, 0);
    attend_combine<<<dim3(D_V / 256),   blk, 0, stream>>>(kk, vv, W_q, b_q, tid, res);
    gemv_wmma<<<dim3(D_T  / 32, 1),     blk, 0, stream>>>(res, W_t,  b_t,   tw,  D_V, D_T,  0,   1);
    gemv_wmma<<<dim3(D_OUT/ 32, 1),     blk, 0, stream>>>(tw,  W_l,  b_l,   out, D_T, D_OUT,0,   0);
}